// MeanAggregator_55284819034566
// MI455X (gfx1250) — compile-verified
//
#include <hip/hip_runtime.h>
#include <hip/hip_bf16.h>

typedef __attribute__((ext_vector_type(2))) float v2f;
typedef __attribute__((ext_vector_type(8))) float v8f;

#define HD   256
#define LSEQ 10
#define ROW  768   // 3*H

// ---------------------------------------------------------------------------
// Kernel 1: zero the [B, L, 3H] output (rows with no history must stay 0).
// ---------------------------------------------------------------------------
__global__ __launch_bounds__(256) void zero_out_kernel(float4* __restrict__ out, int n4) {
    int i = blockIdx.x * blockDim.x + threadIdx.x;
    if (i < n4) out[i] = make_float4(0.f, 0.f, 0.f, 0.f);
}

// ---------------------------------------------------------------------------
// Kernel 2: exclusive prefix sum of len_s -> off[0..T]  (off[T] = N).
// Single block; T ~ 5.6K so this is negligible.
// ---------------------------------------------------------------------------
__global__ __launch_bounds__(256) void scan_kernel(const int* __restrict__ len_s,
                                                   int T, int* __restrict__ off) {
    __shared__ int buf[256];
    __shared__ int carry;
    const int tid = threadIdx.x;
    if (tid == 0) carry = 0;
    __syncthreads();
    for (int base = 0; base < T; base += 256) {
        int i = base + tid;
        int v = (i < T) ? len_s[i] : 0;
        buf[tid] = v;
        __syncthreads();
        #pragma unroll
        for (int s = 1; s < 256; s <<= 1) {
            int add = (tid >= s) ? buf[tid - s] : 0;
            __syncthreads();
            buf[tid] += add;
            __syncthreads();
        }
        int incl = buf[tid];
        if (i < T) off[i] = carry + incl - v;   // exclusive
        __syncthreads();
        if (tid == 255) carry += incl;          // chunk total
        __syncthreads();
    }
    if (tid == 0) off[T] = carry;
}

// ---------------------------------------------------------------------------
// Kernel 3: segment-mean as a masked GEMM on V_WMMA_F32_16X16X4_F32.
// One block = 16 segments. 4 waves, each owning 64 H-columns (4 16x16 tiles).
// A-mask is computed branchlessly from hoisted per-lane [start,end) offsets:
// neighbor j belongs to segment s  <=>  off[s] <= j < off[s+1].
// ---------------------------------------------------------------------------
__global__ __launch_bounds__(128) void seg_mean_wmma_kernel(
    const float* __restrict__ ent,
    const int*   __restrict__ flat_s,
    const int*   __restrict__ len_s,
    const int*   __restrict__ seg_batch,
    const int*   __restrict__ seg_pos,
    const int*   __restrict__ off,
    int T, int N,
    float* __restrict__ out)
{
    const int t0     = blockIdx.x * 16;           // first segment of tile
    const int wave   = threadIdx.x >> 5;          // 0..3
    const int lane   = threadIdx.x & 31;
    const int laneHi = lane >> 4;                 // 0 | 1
    const int l16    = lane & 15;
    const int hbase  = wave * 64;                 // this wave's H-columns

    const int tEnd = (t0 + 16 < T) ? (t0 + 16) : T;
    const int off0 = off[t0];
    const int K    = off[tEnd] - off0;            // contiguous neighbor range

    // Per-lane mask row bounds (hoisted out of the K loop; branchless).
    const int seg   = t0 + l16;                   // this lane's segment row (M)
    const int segc  = (seg < T) ? seg : (T - 1);
    const int sA    = off[segc];
    const int eA    = (seg < T) ? off[segc + 1] : sA;   // empty range if invalid

    const int Nm1 = N - 1;
    const int kk0 = laneHi * 2;                   // K-pair handled by this half-wave

    v8f c0 = {}, c1 = {}, c2 = {}, c3 = {};

    #pragma unroll 2
    for (int k = 0; k < K; k += 4) {
        const int idx0 = off0 + k + kk0;
        const int idx1 = idx0 + 1;
        const int j0 = (idx0 < Nm1) ? idx0 : Nm1;       // safe gather index
        const int j1 = (idx1 < Nm1) ? idx1 : Nm1;

        // ---- A: 16x4 0/1 mask from integer range test (no loads, no branches)
        v2f A;
        A.x = ((idx0 >= sA) & (idx0 < eA)) ? 1.0f : 0.0f;
        A.y = ((idx1 >= sA) & (idx1 < eA)) ? 1.0f : 0.0f;

        // ---- B: 4x16 slabs of the two neighbor rows (coalesced 64B / 16 lanes)
        const float* r0 = ent + (long)flat_s[j0] * HD;
        const float* r1 = ent + (long)flat_s[j1] * HD;
        v2f B0, B1, B2, B3;
        B0.x = r0[hbase +  0 + l16]; B0.y = r1[hbase +  0 + l16];
        B1.x = r0[hbase + 16 + l16]; B1.y = r1[hbase + 16 + l16];
        B2.x = r0[hbase + 32 + l16]; B2.y = r1[hbase + 32 + l16];
        B3.x = r0[hbase + 48 + l16]; B3.y = r1[hbase + 48 + l16];

        c0 = __builtin_amdgcn_wmma_f32_16x16x4_f32(false, A, false, B0, (short)0, c0, false, false);
        c1 = __builtin_amdgcn_wmma_f32_16x16x4_f32(false, A, false, B1, (short)0, c1, false, false);
        c2 = __builtin_amdgcn_wmma_f32_16x16x4_f32(false, A, false, B2, (short)0, c2, false, false);
        c3 = __builtin_amdgcn_wmma_f32_16x16x4_f32(false, A, false, B3, (short)0, c3, false, false);
    }

    // ---- divide by len and scatter rows. D layout: M = v + 8*laneHi, N = l16 ----
    #pragma unroll
    for (int v = 0; v < 8; ++v) {
        const int t = t0 + v + laneHi * 8;
        if (t < T) {
            const float invlen = 1.0f / (float)len_s[t];
            const long base = ((long)seg_batch[t] * LSEQ + seg_pos[t]) * ROW + hbase + l16;
            out[base +  0] = c0[v] * invlen;
            out[base + 16] = c1[v] * invlen;
            out[base + 32] = c2[v] * invlen;
            out[base + 48] = c3[v] * invlen;
        }
    }
}

// ---------------------------------------------------------------------------
// Kernel 4: per segment row, copy ent[s_tem[b]] -> [256:512) and
// rel[r_tem[b]] -> [512:768). One block per segment; float4, fully coalesced.
// ---------------------------------------------------------------------------
__global__ __launch_bounds__(128) void copy_sr_kernel(
    const float* __restrict__ ent, const float* __restrict__ rel,
    const int* __restrict__ s_tem, const int* __restrict__ r_tem,
    const int* __restrict__ seg_batch, const int* __restrict__ seg_pos,
    float* __restrict__ out)
{
    const int t = blockIdx.x;
    const int b = seg_batch[t];
    const long base = ((long)b * LSEQ + seg_pos[t]) * ROW;
    const int tid = threadIdx.x;

    if (tid < 64) {
        const float4* src = (const float4*)(ent + (long)s_tem[b] * HD) + tid;
        *(float4*)(out + base + HD + tid * 4) = *src;
    } else {
        const int q = tid - 64;
        const float4* src = (const float4*)(rel + (long)r_tem[b] * HD) + q;
        *(float4*)(out + base + 2 * HD + q * 4) = *src;
    }
}

// ---------------------------------------------------------------------------
extern "C" void kernel_launch(void* const* d_in, const int* in_sizes, int n_in,
                              void* d_out, int out_size, void* d_ws, size_t ws_size,
                              hipStream_t stream) {
    const float* ent       = (const float*)d_in[0];
    const float* rel       = (const float*)d_in[1];
    const int*   flat_s    = (const int*)d_in[2];
    // d_in[3] = seg_ids: no longer needed (mask derived from offsets)
    const int*   len_s     = (const int*)d_in[4];
    const int*   seg_batch = (const int*)d_in[5];
    const int*   seg_pos   = (const int*)d_in[6];
    const int*   s_tem     = (const int*)d_in[7];
    const int*   r_tem     = (const int*)d_in[8];

    const int T = in_sizes[4];     // number of history segments
    const int N = in_sizes[2];     // total flat neighbors
    float* out = (float*)d_out;
    int*   off = (int*)d_ws;       // T+1 ints of scratch

    // 1) zero-fill output
    const int n4 = out_size / 4;
    zero_out_kernel<<<(n4 + 255) / 256, 256, 0, stream>>>((float4*)out, n4);

    // 2) exclusive prefix sum of len_s
    scan_kernel<<<1, 256, 0, stream>>>(len_s, T, off);

    // 3) WMMA masked segment-mean + scatter
    const int ntiles = (T + 15) / 16;
    seg_mean_wmma_kernel<<<ntiles, 128, 0, stream>>>(
        ent, flat_s, len_s, seg_batch, seg_pos, off, T, N, out);

    // 4) s/r embedding sections
    copy_sr_kernel<<<T, 128, 0, stream>>>(ent, rel, s_tem, r_tem, seg_batch, seg_pos, out);
}